// GenProcessFinal_11295763989053
// MI455X (gfx1250) — compile-verified
//
#include <hip/hip_runtime.h>

// Problem shape (fixed by the reference): x[64,2048,256] f32, W[256,512] f32 -> out[64,512] f32.
// Cost model: streaming x (134 MB) dominates -> ~5.8us at 23.3 TB/s. GEMM after the
// seq-mean is 8.4 MFLOP (negligible) and is done with V_WMMA_F32_16X16X4_F32.

namespace {
constexpr int Bn = 64;
constexpr int Sn = 2048;
constexpr int Kn = 256;
constexpr int Dn = 512;
constexpr int CHUNKS = 32;              // s-chunks per batch for the partial reduction
constexpr int S_CHUNK = Sn / CHUNKS;    // 64
}

typedef float v2f __attribute__((ext_vector_type(2)));
typedef float v8f __attribute__((ext_vector_type(8)));

// ---------------- Phase 1: partial sum over a chunk of S (bandwidth-bound) -------------
// grid = (CHUNKS, Bn), block = 64. Thread t owns float4 lane k4=t; streams 64 s-rows.
__global__ __launch_bounds__(64)
void seq_partial_kernel(const float* __restrict__ x, float* __restrict__ partial) {
    const int c = blockIdx.x;
    const int b = blockIdx.y;
    const int t = threadIdx.x;                     // 0..63  (float4 index over K)
    const float4* __restrict__ xr =
        reinterpret_cast<const float4*>(x) +
        ((size_t)b * Sn + (size_t)c * S_CHUNK) * (Kn / 4) + t;

    float4 a0 = make_float4(0.f, 0.f, 0.f, 0.f);
    float4 a1 = a0, a2 = a0, a3 = a0;
    #pragma unroll 4
    for (int s = 0; s < S_CHUNK; s += 4) {
        float4 r0 = xr[(size_t)(s + 0) * (Kn / 4)];
        float4 r1 = xr[(size_t)(s + 1) * (Kn / 4)];
        float4 r2 = xr[(size_t)(s + 2) * (Kn / 4)];
        float4 r3 = xr[(size_t)(s + 3) * (Kn / 4)];
        a0.x += r0.x; a0.y += r0.y; a0.z += r0.z; a0.w += r0.w;
        a1.x += r1.x; a1.y += r1.y; a1.z += r1.z; a1.w += r1.w;
        a2.x += r2.x; a2.y += r2.y; a2.z += r2.z; a2.w += r2.w;
        a3.x += r3.x; a3.y += r3.y; a3.z += r3.z; a3.w += r3.w;
    }
    float4 acc;
    acc.x = (a0.x + a1.x) + (a2.x + a3.x);
    acc.y = (a0.y + a1.y) + (a2.y + a3.y);
    acc.z = (a0.z + a1.z) + (a2.z + a3.z);
    acc.w = (a0.w + a1.w) + (a2.w + a3.w);

    float4* __restrict__ p4 =
        reinterpret_cast<float4*>(partial) + ((size_t)(b * CHUNKS + c)) * (Kn / 4) + t;
    *p4 = acc;
}

// ---------------- Phase 2: combine partials, scale by 1/S ------------------------------
// grid = Bn, block = Kn(=256). Thread k sums 32 partials (coalesced across k).
__global__ __launch_bounds__(256)
void mean_finalize_kernel(const float* __restrict__ partial, float* __restrict__ xmean) {
    const int b = blockIdx.x;
    const int k = threadIdx.x;
    float s = 0.f;
    #pragma unroll
    for (int c = 0; c < CHUNKS; ++c)
        s += partial[((size_t)(b * CHUNKS + c)) * Kn + k];
    xmean[(size_t)b * Kn + k] = s * (1.0f / (float)Sn);
}

// ---------------- Phase 3: [64,256] x [256,512] GEMM via V_WMMA_F32_16X16X4_F32 --------
// One 16x16 output tile per wave32. 4 M-tiles * 32 N-tiles = 128 waves = 16 blocks of 256.
__global__ __launch_bounds__(256)
void wmma_gemm_kernel(const float* __restrict__ xmean, const float* __restrict__ W,
                      float* __restrict__ out) {
    const int lane = threadIdx.x & 31;
    const int wave = (blockIdx.x << 3) | (threadIdx.x >> 5);  // 0..127
    const int mt   = wave >> 5;                               // 0..3
    const int nt   = wave & 31;                               // 0..31
    const int m0   = mt << 4;
    const int n0   = nt << 4;
    const int lh   = lane >> 4;                               // lane half: 0 or 1
    const int li   = lane & 15;

    v8f acc = {0.f, 0.f, 0.f, 0.f, 0.f, 0.f, 0.f, 0.f};

    // A (16x4 f32): lanes 0-15 hold M=li with K={k0,k0+1}; lanes 16-31 hold K={k0+2,k0+3}.
    const float* __restrict__ arow = xmean + (size_t)(m0 + li) * Kn;
    // B (4x16 f32): VGPR0 = rows {k0, k0+2} split across lane halves, VGPR1 = {k0+1, k0+3}.
    const float* __restrict__ bcol = W + n0 + li;

    #pragma unroll 8
    for (int k0 = 0; k0 < Kn; k0 += 4) {
        const int kA = k0 + (lh << 1);
        v2f a = *reinterpret_cast<const v2f*>(arow + kA);   // 8B-aligned (kA even)
        v2f bm;
        bm.x = bcol[(size_t)kA * Dn];
        bm.y = bcol[(size_t)(kA + 1) * Dn];
        acc = __builtin_amdgcn_wmma_f32_16x16x4_f32(
            /*neg_a=*/false, a, /*neg_b=*/false, bm,
            /*c_mod=*/(short)0, acc, /*reuse_a=*/false, /*reuse_b=*/false);
    }

    // C/D layout: VGPR v -> row m0 + lh*8 + v, col n0 + li.
    float* __restrict__ orow = out + (size_t)(m0 + (lh << 3)) * Dn + n0 + li;
    #pragma unroll
    for (int v = 0; v < 8; ++v)
        orow[(size_t)v * Dn] = acc[v];
}

extern "C" void kernel_launch(void* const* d_in, const int* in_sizes, int n_in,
                              void* d_out, int out_size, void* d_ws, size_t ws_size,
                              hipStream_t stream) {
    const float* x = (const float*)d_in[0];     // [64, 2048, 256]
    const float* W = (const float*)d_in[1];     // [256, 512]
    float* out = (float*)d_out;                 // [64, 512]

    float* partial = (float*)d_ws;                          // Bn*CHUNKS*Kn floats (2 MB)
    float* xmean   = partial + (size_t)Bn * CHUNKS * Kn;    // Bn*Kn floats (64 KB)

    seq_partial_kernel<<<dim3(CHUNKS, Bn), 64, 0, stream>>>(x, partial);
    mean_finalize_kernel<<<Bn, Kn, 0, stream>>>(partial, xmean);
    wmma_gemm_kernel<<<(Bn / 16) * (Dn / 16) / 8, 256, 0, stream>>>(xmean, W, out);
}